// Step3p5Router_6674379178225
// MI455X (gfx1250) — compile-verified
//
#include <hip/hip_runtime.h>

typedef __attribute__((ext_vector_type(2))) float        v2f;
typedef __attribute__((ext_vector_type(8))) float        v8f;
typedef __attribute__((ext_vector_type(4))) unsigned int u32x4;
typedef __attribute__((ext_vector_type(4))) int          i32x4;
typedef __attribute__((ext_vector_type(8))) int          i32x8;

#define H_DIM   4096
#define E_DIM   64
#define TOPK    8
#define KC      64          // K-chunk staged in LDS
#define LDA     (KC + 4)    // padded row stride (floats): TDM pad 4 DWORDs every 64 DWORDs
#define TPB     128         // tokens per block
#define THREADS 256         // 8 waves
#define LDP     (E_DIM + 1) // prob row stride: 65 -> conflict-free row scans
#define SCALE   2.5f
#define EPSV    1e-20f

#define XBUF_FLOATS (TPB * LDA)                      // 8704 floats / buffer
#define WBUF_FLOATS (E_DIM * LDA)                    // 4352 floats / buffer
#define WOFF_FLOATS (2 * XBUF_FLOATS)                // 17408
#define BIAS_OFF    (WOFF_FLOATS + 2 * WBUF_FLOATS)  // 26112
#define SMEM_FLOATS (BIAS_OFF + E_DIM)
#define SMEM_BYTES  (SMEM_FLOATS * 4)                // 104704 B < 320 KB/WGP

__device__ __forceinline__ float sigmoidf(float x) {
    return 1.0f / (1.0f + __expf(-x));
}

// TDM: DMA a 2D f32 tile [tile_h rows x tile_w elems] (row stride = stride elems)
// from global to LDS, inserting 4 pad DWORDs after every 64 DWORDs (-> LDA=68 rows).
// Descriptor packing per CDNA5 ISA section 8.3/8.4 (D# groups 0/1; groups 2/3 zero for 2D).
__device__ __forceinline__ void tdm_load_2d_f32(
    const float* gsrc, unsigned lds_byte, int tile_w, int tile_h, int stride)
{
    unsigned long long ga = (unsigned long long)(size_t)gsrc;

    u32x4 g0;
    g0[0] = 1u;                                           // count=1 (valid), user mode
    g0[1] = lds_byte;                                     // lds_addr (bytes)
    g0[2] = (unsigned)ga;                                 // global_addr[31:0]
    g0[3] = (unsigned)((ga >> 32) & 0x01FFFFFFull)        // global_addr[56:32]
          | (2u << 30);                                   // type=2 ("image")

    const unsigned td0 = (unsigned)stride;                // tensor_dim0 (>= tile_w; no OOB)
    const unsigned td1 = 1u << 20;                        // tensor_dim1 (>= tile_h; no OOB)
    i32x8 g1;
    g1[0] = (int)((2u << 16)                              // data_size = 4 bytes
                | (1u << 20)                              // pad_enable
                | (5u << 22)                              // pad_interval: 2<<5 = 64 DWORDs
                | (3u << 25));                            // pad_amount: 3+1 = 4 DWORDs
    g1[1] = (int)((td0 & 0xFFFFu) << 16);                 // tensor_dim0[15:0]
    g1[2] = (int)(((td0 >> 16) & 0xFFFFu)                 // tensor_dim0[31:16]
                | ((td1 & 0xFFFFu) << 16));               // tensor_dim1[15:0]
    g1[3] = (int)(((td1 >> 16) & 0xFFFFu)                 // tensor_dim1[31:16]
                | (((unsigned)tile_w & 0xFFFFu) << 16));  // tile_dim0
    g1[4] = (int)((unsigned)tile_h & 0xFFFFu);            // tile_dim1 (tile_dim2 = 0)
    g1[5] = stride;                                       // tensor_dim0_stride[31:0]
    g1[6] = 0;                                            // stride[47:32] | dim1_stride lo (unused, 2D)
    g1[7] = 0;

    i32x4 gz = {0, 0, 0, 0};
#if __has_include(<hip/amd_detail/amd_gfx1250_TDM.h>)
    // therock / clang-23 headers present -> 6-arg builtin
    i32x8 gz8 = {0, 0, 0, 0, 0, 0, 0, 0};
    __builtin_amdgcn_tensor_load_to_lds(g0, g1, gz, gz, gz8, 0);
#else
    // ROCm 7.2 / clang-22 -> 5-arg builtin
    __builtin_amdgcn_tensor_load_to_lds(g0, g1, gz, gz, 0);
#endif
}

__global__ __launch_bounds__(THREADS) void step35_router_kernel(
    const float* __restrict__ x,     // [T, H]
    const float* __restrict__ gw,    // [E, H]
    const float* __restrict__ bias,  // [E]
    float* __restrict__ outW,        // [T, 8]
    int*   __restrict__ outI)        // [T, 8]
{
    extern __shared__ float smem[];  // [X0 | X1 | W0 | W1 | bias]; probs reuse X0

    const int t    = threadIdx.x;
    const int wave = t >> 5;
    const int lane = t & 31;
    const int l15  = lane & 15;
    const int hi   = lane >> 4;
    const int tok0 = blockIdx.x * TPB;

    float* ldsBias = smem + BIAS_OFF;
    if (t < E_DIM) ldsBias[t] = bias[t];

    const unsigned smem_base = (unsigned)(size_t)(void*)smem;  // LDS byte offset of dyn base
    const float* xBlk = x + (size_t)tok0 * H_DIM;

    // ---- prologue: DMA chunk 0 into buffer 0 ----
    if (wave == 0) {
        tdm_load_2d_f32(xBlk,  smem_base,                     KC, TPB,   H_DIM);
        tdm_load_2d_f32(gw,    smem_base + WOFF_FLOATS * 4u,  KC, E_DIM, H_DIM);
        __builtin_amdgcn_s_wait_tensorcnt(0);
    }
    __syncthreads();

    v8f c0 = {}, c1 = {}, c2 = {}, c3 = {};

    for (int c = 0; c < H_DIM / KC; ++c) {
        // ---- issue DMA for chunk c+1 into the other buffer (overlaps compute) ----
        const int k0n = (c + 1) * KC;
        if (wave == 0 && k0n < H_DIM) {
            const unsigned pb = (unsigned)((c + 1) & 1);
            tdm_load_2d_f32(xBlk + k0n,
                            smem_base + pb * (XBUF_FLOATS * 4u), KC, TPB, H_DIM);
            tdm_load_2d_f32(gw + k0n,
                            smem_base + (WOFF_FLOATS + pb * WBUF_FLOATS) * 4u, KC, E_DIM, H_DIM);
        }

        // ---- compute on current buffer ----
        const float* Xb = smem + (c & 1) * XBUF_FLOATS;
        const float* Wb = smem + WOFF_FLOATS + (c & 1) * WBUF_FLOATS;

        const float* aRow  = &Xb[(wave * 16 + l15) * LDA + 2 * hi];
        const float* b0Row = &Wb[( 0 + l15) * LDA + 2 * hi];
        const float* b1Row = &Wb[(16 + l15) * LDA + 2 * hi];
        const float* b2Row = &Wb[(32 + l15) * LDA + 2 * hi];
        const float* b3Row = &Wb[(48 + l15) * LDA + 2 * hi];

        #pragma unroll
        for (int kk = 0; kk < KC; kk += 4) {
            v2f a  = *(const v2f*)(aRow  + kk);
            v2f b0 = *(const v2f*)(b0Row + kk);
            v2f b1 = *(const v2f*)(b1Row + kk);
            v2f b2 = *(const v2f*)(b2Row + kk);
            v2f b3 = *(const v2f*)(b3Row + kk);
            c0 = __builtin_amdgcn_wmma_f32_16x16x4_f32(false, a, false, b0, (short)0, c0, false, false);
            c1 = __builtin_amdgcn_wmma_f32_16x16x4_f32(false, a, false, b1, (short)0, c1, false, false);
            c2 = __builtin_amdgcn_wmma_f32_16x16x4_f32(false, a, false, b2, (short)0, c2, false, false);
            c3 = __builtin_amdgcn_wmma_f32_16x16x4_f32(false, a, false, b3, (short)0, c3, false, false);
        }

        // ---- close the chunk: next buffer DMA complete + all waves done reading ----
        if (wave == 0) __builtin_amdgcn_s_wait_tensorcnt(0);
        __syncthreads();
    }

    // ---- epilogue: sigmoid(logits) -> prob LDS (reuse X0 region) ----
    // C layout: VGPR j: lanes 0-15 -> M=j, N=lane; lanes 16-31 -> M=j+8, N=lane-16
    float* probs = smem;  // [TPB][LDP] = 33280 B <= X0 buffer (34816 B)
    const int tokBase = wave * 16 + hi * 8;
    #pragma unroll
    for (int j = 0; j < 8; ++j) {
        float* row = &probs[(tokBase + j) * LDP + l15];
        row[ 0] = sigmoidf(c0[j]);
        row[16] = sigmoidf(c1[j]);
        row[32] = sigmoidf(c2[j]);
        row[48] = sigmoidf(c3[j]);
    }
    __syncthreads();

    // ---- top-8 on biased probs; weights from unbiased probs ----
    if (t < TPB) {
        const float* pr = &probs[t * LDP];
        unsigned long long mask = 0ull;
        float sel[TOPK];
        int   idx[TOPK];
        float psum = 0.0f;
        for (int i = 0; i < TOPK; ++i) {
            float best = -3.4e38f;
            int   bi   = 0;
            for (int e = 0; e < E_DIM; ++e) {
                bool taken = (mask >> e) & 1ull;
                float bv = pr[e] + ldsBias[e];
                if (!taken && bv > best) { best = bv; bi = e; }  // strict > => lowest index on ties
            }
            mask |= (1ull << bi);
            float p = pr[bi];
            sel[i] = p;
            idx[i] = bi;
            psum  += p;
        }
        float inv = SCALE / (psum + EPSV);
        const int gtok = tok0 + t;
        #pragma unroll
        for (int i = 0; i < TOPK; ++i) {
            outW[(size_t)gtok * TOPK + i] = sel[i] * inv;
            outI[(size_t)gtok * TOPK + i] = idx[i];
        }
    }
}

extern "C" void kernel_launch(void* const* d_in, const int* in_sizes, int n_in,
                              void* d_out, int out_size, void* d_ws, size_t ws_size,
                              hipStream_t stream) {
    const float* x    = (const float*)d_in[0];   // hidden_states [B,S,H] fp32
    const float* gw   = (const float*)d_in[1];   // gate_weight   [E,H]   fp32
    const float* bias = (const float*)d_in[2];   // router_bias   [E]     fp32

    const int T = in_sizes[0] / H_DIM;           // 16384 tokens

    float* outW = (float*)d_out;                 // weights [T,8] fp32
    int*   outI = (int*)d_out + (size_t)T * TOPK;// indices [T,8] int32 (concatenated)

    // Allow >64KB dynamic LDS (320 KB per WGP on gfx1250); ignore failure.
    (void)hipFuncSetAttribute((const void*)step35_router_kernel,
                              hipFuncAttributeMaxDynamicSharedMemorySize, SMEM_BYTES);

    dim3 grid(T / TPB);
    step35_router_kernel<<<grid, THREADS, SMEM_BYTES, stream>>>(x, gw, bias, outW, outI);
}